// ImprovedNewsGNN_25649544691770
// MI455X (gfx1250) — compile-verified
//
#include <hip/hip_runtime.h>
#include <hip/hip_bf16.h>
#include <math.h>

#define HID 128
#define HEADS 4
#define HC 32
#define LN_EPS 1e-5f

typedef __attribute__((ext_vector_type(2))) float v2f;
typedef __attribute__((ext_vector_type(8))) float v8f;

// ---------------------------------------------------------------------------
// f32 WMMA helper: D = A(16x4) * B(4x16) + C, full f32 precision.
// A frag (2 VGPR): lanes 0-15 hold M=lane,    K = k+0 (x), k+1 (y)
//                  lanes 16-31 hold M=lane-16, K = k+2 (x), k+3 (y)
// B frag (2 VGPR): lanes 0-15 hold N=lane,    K = k+0 (x), k+1 (y)
//                  lanes 16-31 hold N=lane-16, K = k+2 (x), k+3 (y)
// C/D (8 VGPR): vgpr r, lane l -> (M = r + 8*(l>=16), N = l&15)
// ---------------------------------------------------------------------------
__device__ inline v8f wmma_f32(v2f a, v2f b, v8f c) {
    return __builtin_amdgcn_wmma_f32_16x16x4_f32(
        false, a, false, b, (short)0, c, false, false);
}

// ---------------------------------------------------------------------------
// Fused GEMM (nrows x K @ K x NCOLS) + bias + LayerNorm + ReLU (+ type emb)
// Block: NCOLS/16 waves, each wave owns one 16-col tile; 16 rows per block.
// A tile staged once into LDS (row stride K+4 -> bank-conflict-free frags).
// ---------------------------------------------------------------------------
template <int NCOLS, int K>
__global__ __launch_bounds__(32 * (NCOLS / 16))
void gemm_ln_relu_kernel(const float* __restrict__ X,
                         const float* __restrict__ W,
                         const float* __restrict__ bias,
                         const float* __restrict__ ln_g,
                         const float* __restrict__ ln_b,
                         const float* __restrict__ type_emb,  // may be null
                         float* __restrict__ out,
                         int nrows) {
    constexpr int NT = NCOLS / 16;
    constexpr int BS = 32 * NT;
    constexpr int KP = K + 4;  // padded LDS row stride (banks 0,4,..,60)
    __shared__ float As[16 * KP];
    __shared__ float tile[16][NCOLS + 4];
    __shared__ float smu[16], srs[16];

    const int tid  = threadIdx.x;
    const int wave = tid >> 5;
    const int lane = tid & 31;
    const int m0   = blockIdx.x * 16;
    const int n0   = wave * 16;
    const int half = lane >> 4;   // 0: K 0/1, 1: K 2/3
    const int l15  = lane & 15;

    // cooperative stage of the 16 x K A tile (b128 loads / stores)
    for (int idx = tid; idx < 16 * (K / 4); idx += BS) {
        int r   = idx / (K / 4);
        int c4  = idx - r * (K / 4);
        int row = m0 + r;
        if (row >= nrows) row = nrows - 1;
        const float4 v = *(const float4*)(X + (size_t)row * K + c4 * 4);
        *(float4*)(As + r * KP + c4 * 4) = v;
    }
    __syncthreads();

    const float* Arow = As + l15 * KP + half * 2;
    const float* Bcol = W + n0 + l15 + (size_t)(half * 2) * NCOLS;

    v8f acc = {};
#pragma unroll 8
    for (int k = 0; k < K; k += 4) {
        v2f a, b;
        a.x = Arow[k];
        a.y = Arow[k + 1];
        b.x = Bcol[(size_t)k * NCOLS];
        b.y = Bcol[(size_t)(k + 1) * NCOLS];
        acc = wmma_f32(a, b, acc);
    }
    __syncthreads();
#pragma unroll
    for (int r = 0; r < 8; ++r) {
        int m = r + half * 8;
        int n = n0 + l15;
        tile[m][n] = acc[r] + bias[n];
    }
    __syncthreads();
    if (tid < 16) {
        float s = 0.f, s2 = 0.f;
        for (int n = 0; n < NCOLS; ++n) {
            float v = tile[tid][n];
            s += v; s2 += v * v;
        }
        float mu  = s / NCOLS;
        float var = s2 / NCOLS - mu * mu;
        smu[tid] = mu;
        srs[tid] = rsqrtf(var + LN_EPS);
    }
    __syncthreads();
    for (int idx = tid; idx < 16 * NCOLS; idx += BS) {
        int m = idx / NCOLS, n = idx % NCOLS;
        int row = m0 + m;
        if (row < nrows) {
            float v = (tile[m][n] - smu[m]) * srs[m] * ln_g[n] + ln_b[n];
            v = v > 0.f ? v : 0.f;
            if (type_emb) v += type_emb[n];
            out[(size_t)row * NCOLS + n] = v;
        }
    }
}

// ---------------------------------------------------------------------------
// GAT: H = X @ W (128x128), plus per-head attention dots a_src, a_dst.
// ---------------------------------------------------------------------------
__global__ __launch_bounds__(256)
void gat_h_kernel(const float* __restrict__ X, const float* __restrict__ W,
                  const float* __restrict__ att_src,
                  const float* __restrict__ att_dst,
                  float* __restrict__ H, float* __restrict__ Asrc,
                  float* __restrict__ Adst, int nrows) {
    constexpr int KP = HID + 4;
    __shared__ float As[16 * KP];
    __shared__ float tile[16][HID + 4];
    const int tid  = threadIdx.x;
    const int wave = tid >> 5;
    const int lane = tid & 31;
    const int m0   = blockIdx.x * 16;
    const int n0   = wave * 16;
    const int half = lane >> 4;
    const int l15  = lane & 15;

    for (int idx = tid; idx < 16 * (HID / 4); idx += 256) {
        int r   = idx >> 5;          // HID/4 == 32
        int c4  = idx & 31;
        int row = m0 + r;
        if (row >= nrows) row = nrows - 1;
        const float4 v = *(const float4*)(X + (size_t)row * HID + c4 * 4);
        *(float4*)(As + r * KP + c4 * 4) = v;
    }
    __syncthreads();

    const float* Arow = As + l15 * KP + half * 2;
    const float* Bcol = W + n0 + l15 + (size_t)(half * 2) * HID;

    v8f acc = {};
#pragma unroll 8
    for (int k = 0; k < HID; k += 4) {
        v2f a, b;
        a.x = Arow[k];
        a.y = Arow[k + 1];
        b.x = Bcol[(size_t)k * HID];
        b.y = Bcol[(size_t)(k + 1) * HID];
        acc = wmma_f32(a, b, acc);
    }
    __syncthreads();
#pragma unroll
    for (int r = 0; r < 8; ++r) tile[r + half * 8][n0 + l15] = acc[r];
    __syncthreads();
    for (int idx = tid; idx < 16 * HID; idx += 256) {
        int m = idx >> 7, n = idx & (HID - 1);
        int row = m0 + m;
        if (row < nrows) H[(size_t)row * HID + n] = tile[m][n];
    }
    if (tid < 16 * HEADS) {
        int m = tid >> 2, hh = tid & 3;
        int row = m0 + m;
        float s1 = 0.f, s2 = 0.f;
        for (int c = 0; c < HC; ++c) {
            float v = tile[m][hh * HC + c];
            s1 += v * att_src[hh * HC + c];
            s2 += v * att_dst[hh * HC + c];
        }
        if (row < nrows) {
            Asrc[(size_t)row * HEADS + hh] = s1;
            Adst[(size_t)row * HEADS + hh] = s2;
        }
    }
}

// ---------------------------------------------------------------------------
// Edge softmax + aggregation
// ---------------------------------------------------------------------------
__device__ inline void edge_sd(const int* __restrict__ ei, int e, int E,
                               int& s, int& d) {
    if (e < E) { s = ei[e]; d = ei[E + e]; }
    else       { s = e - E; d = s; }          // self loops appended
}

__device__ inline void atomicMaxFloat(float* addr, float val) {
    int* ia = (int*)addr;
    int cur = *ia;
    while (__int_as_float(cur) < val) {
        int prev = atomicCAS(ia, cur, __float_as_int(val));
        if (prev == cur) break;
        cur = prev;
    }
}

__global__ void init_kernel(float* __restrict__ M, float* __restrict__ D,
                            float* __restrict__ OUTB, int n) {
    size_t idx = (size_t)blockIdx.x * blockDim.x + threadIdx.x;
    if (idx < (size_t)n * HID) OUTB[idx] = 0.f;
    if (idx < (size_t)n * HEADS) { M[idx] = -3.4e38f; D[idx] = 0.f; }
}

__global__ void edge_max_kernel(const int* __restrict__ ei,
                                const float* __restrict__ As,
                                const float* __restrict__ Ad,
                                float* __restrict__ M, int E, int Nn) {
    int e = blockIdx.x * blockDim.x + threadIdx.x;
    if (e >= E + Nn) return;
    int s, d; edge_sd(ei, e, E, s, d);
#pragma unroll
    for (int h = 0; h < HEADS; ++h) {
        float v = As[(size_t)s * HEADS + h] + Ad[(size_t)d * HEADS + h];
        v = v > 0.f ? v : 0.2f * v;  // leaky_relu(0.2)
        atomicMaxFloat(&M[(size_t)d * HEADS + h], v);
    }
}

__global__ void edge_expsum_kernel(const int* __restrict__ ei,
                                   const float* __restrict__ As,
                                   const float* __restrict__ Ad,
                                   const float* __restrict__ M,
                                   float* __restrict__ EXB,
                                   float* __restrict__ D, int E, int Nn) {
    int e = blockIdx.x * blockDim.x + threadIdx.x;
    if (e >= E + Nn) return;
    int s, d; edge_sd(ei, e, E, s, d);
#pragma unroll
    for (int h = 0; h < HEADS; ++h) {
        float v = As[(size_t)s * HEADS + h] + Ad[(size_t)d * HEADS + h];
        v = v > 0.f ? v : 0.2f * v;
        float ex = expf(v - M[(size_t)d * HEADS + h]);
        EXB[(size_t)e * HEADS + h] = ex;
        atomicAdd(&D[(size_t)d * HEADS + h], ex);
    }
}

// one wave per edge; lane = feature within 32-wide head chunk
__global__ void edge_msg_kernel(const int* __restrict__ ei,
                                const float* __restrict__ H,
                                const float* __restrict__ EXB,
                                const float* __restrict__ Dn,
                                float* __restrict__ OUTB, int E, int Nn) {
    size_t gid = (size_t)blockIdx.x * blockDim.x + threadIdx.x;
    int e    = (int)(gid >> 5);
    int lane = (int)(gid & 31);
    if (e >= E + Nn) return;
    int s, d; edge_sd(ei, e, E, s, d);
#pragma unroll
    for (int j = 0; j < HEADS; ++j) {
        float alpha = EXB[(size_t)e * HEADS + j] / Dn[(size_t)d * HEADS + j];
        int f = j * 32 + lane;
        atomicAdd(&OUTB[(size_t)d * HID + f], H[(size_t)s * HID + f] * alpha);
    }
}

// out + bias -> elu -> + residual(X) -> layernorm -> X   (one wave per node)
__global__ void post_gat_kernel(const float* __restrict__ OUTB,
                                const float* __restrict__ bias,
                                const float* __restrict__ g,
                                const float* __restrict__ b,
                                float* __restrict__ X, int Nn) {
    int t    = blockIdx.x * blockDim.x + threadIdx.x;
    int node = t >> 5;
    int lane = t & 31;
    if (node >= Nn) return;
    float v[4];
    float s = 0.f;
#pragma unroll
    for (int j = 0; j < 4; ++j) {
        int f = j * 32 + lane;
        float x = OUTB[(size_t)node * HID + f] + bias[f];
        x = x > 0.f ? x : (expf(x) - 1.f);          // elu
        x += X[(size_t)node * HID + f];             // residual
        v[j] = x; s += x;
    }
#pragma unroll
    for (int o = 16; o > 0; o >>= 1) s += __shfl_xor(s, o);
    float mu  = s * (1.f / HID);
    float var = 0.f;
#pragma unroll
    for (int j = 0; j < 4; ++j) { float dd = v[j] - mu; var += dd * dd; }
#pragma unroll
    for (int o = 16; o > 0; o >>= 1) var += __shfl_xor(var, o);
    var *= (1.f / HID);
    float rs = rsqrtf(var + LN_EPS);
#pragma unroll
    for (int j = 0; j < 4; ++j) {
        int f = j * 32 + lane;
        X[(size_t)node * HID + f] = (v[j] - mu) * rs * g[f] + b[f];
    }
}

// final tiny GEMM: (n x 64) @ (64 x 2) + b3
__global__ void cls3_kernel(const float* __restrict__ Z,
                            const float* __restrict__ w3,
                            const float* __restrict__ b3,
                            float* __restrict__ out, int n) {
    int i = blockIdx.x * blockDim.x + threadIdx.x;
    if (i >= n) return;
    float a0 = b3[0], a1 = b3[1];
#pragma unroll 8
    for (int c = 0; c < 64; ++c) {
        float z = Z[(size_t)i * 64 + c];
        a0 += z * w3[c * 2 + 0];
        a1 += z * w3[c * 2 + 1];
    }
    out[(size_t)i * 2 + 0] = a0;
    out[(size_t)i * 2 + 1] = a1;
}

// ---------------------------------------------------------------------------
extern "C" void kernel_launch(void* const* d_in, const int* in_sizes, int n_in,
                              void* d_out, int out_size, void* d_ws,
                              size_t ws_size, hipStream_t stream) {
    (void)n_in; (void)out_size; (void)ws_size;
    const float* x_news        = (const float*)d_in[0];
    const float* x_tweets      = (const float*)d_in[1];
    const int*   edge_index    = (const int*)d_in[2];
    const float* news_w        = (const float*)d_in[3];
    const float* news_b        = (const float*)d_in[4];
    const float* news_ln_g     = (const float*)d_in[5];
    const float* news_ln_b     = (const float*)d_in[6];
    const float* tweet_w       = (const float*)d_in[7];
    const float* tweet_b       = (const float*)d_in[8];
    const float* tweet_ln_g    = (const float*)d_in[9];
    const float* tweet_ln_b    = (const float*)d_in[10];
    const float* news_type     = (const float*)d_in[11];
    const float* tweet_type    = (const float*)d_in[12];
    const float* gat_w[2]      = {(const float*)d_in[13], (const float*)d_in[17]};
    const float* gat_as[2]     = {(const float*)d_in[14], (const float*)d_in[18]};
    const float* gat_ad[2]     = {(const float*)d_in[15], (const float*)d_in[19]};
    const float* gat_bias[2]   = {(const float*)d_in[16], (const float*)d_in[20]};
    const float* norm_g[2]     = {(const float*)d_in[21], (const float*)d_in[23]};
    const float* norm_b[2]     = {(const float*)d_in[22], (const float*)d_in[24]};
    const float* cls_w1        = (const float*)d_in[25];
    const float* cls_b1        = (const float*)d_in[26];
    const float* cls_ln1_g     = (const float*)d_in[27];
    const float* cls_ln1_b     = (const float*)d_in[28];
    const float* cls_w2        = (const float*)d_in[29];
    const float* cls_b2        = (const float*)d_in[30];
    const float* cls_ln2_g     = (const float*)d_in[31];
    const float* cls_ln2_b     = (const float*)d_in[32];
    const float* cls_w3        = (const float*)d_in[33];
    const float* cls_b3        = (const float*)d_in[34];

    const int Nnews = in_sizes[0] / 768;
    const int Ntw   = in_sizes[1] / 768;
    const int Nn    = Nnews + Ntw;
    const int E     = in_sizes[2] / 2;
    const int ET    = E + Nn;  // edges + self loops

    float* ws = (float*)d_ws;
    size_t o = 0;
    float* X    = ws + o; o += (size_t)Nn * HID;
    float* Hh   = ws + o; o += (size_t)Nn * HID;
    float* OUTB = ws + o; o += (size_t)Nn * HID;
    float* ASRC = ws + o; o += (size_t)Nn * HEADS;
    float* ADST = ws + o; o += (size_t)Nn * HEADS;
    float* MB   = ws + o; o += (size_t)Nn * HEADS;
    float* DB   = ws + o; o += (size_t)Nn * HEADS;
    float* EXB  = ws + o; o += (size_t)ET * HEADS;
    float* Z1   = Hh;    // classifier scratch reuses H
    float* Z2   = OUTB;  // classifier scratch reuses OUT

    // 1) input projections (fused GEMM+LN+ReLU+type_emb) -> X
    gemm_ln_relu_kernel<HID, 768><<<(Nnews + 15) / 16, 256, 0, stream>>>(
        x_news, news_w, news_b, news_ln_g, news_ln_b, news_type, X, Nnews);
    gemm_ln_relu_kernel<HID, 768><<<(Ntw + 15) / 16, 256, 0, stream>>>(
        x_tweets, tweet_w, tweet_b, tweet_ln_g, tweet_ln_b, tweet_type,
        X + (size_t)Nnews * HID, Ntw);

    // 2) two GAT layers
    for (int L = 0; L < 2; ++L) {
        gat_h_kernel<<<(Nn + 15) / 16, 256, 0, stream>>>(
            X, gat_w[L], gat_as[L], gat_ad[L], Hh, ASRC, ADST, Nn);
        init_kernel<<<((size_t)Nn * HID + 255) / 256, 256, 0, stream>>>(
            MB, DB, OUTB, Nn);
        edge_max_kernel<<<(ET + 255) / 256, 256, 0, stream>>>(
            edge_index, ASRC, ADST, MB, E, Nn);
        edge_expsum_kernel<<<(ET + 255) / 256, 256, 0, stream>>>(
            edge_index, ASRC, ADST, MB, EXB, DB, E, Nn);
        edge_msg_kernel<<<((size_t)ET * 32 + 255) / 256, 256, 0, stream>>>(
            edge_index, Hh, EXB, DB, OUTB, E, Nn);
        post_gat_kernel<<<((size_t)Nn * 32 + 255) / 256, 256, 0, stream>>>(
            OUTB, gat_bias[L], norm_g[L], norm_b[L], X, Nn);
    }

    // 3) classifier on first Nnews rows
    gemm_ln_relu_kernel<HID, HID><<<(Nnews + 15) / 16, 256, 0, stream>>>(
        X, cls_w1, cls_b1, cls_ln1_g, cls_ln1_b, nullptr, Z1, Nnews);
    gemm_ln_relu_kernel<64, HID><<<(Nnews + 15) / 16, 128, 0, stream>>>(
        Z1, cls_w2, cls_b2, cls_ln2_g, cls_ln2_b, nullptr, Z2, Nnews);
    cls3_kernel<<<(Nnews + 255) / 256, 256, 0, stream>>>(
        Z2, cls_w3, cls_b3, (float*)d_out, Nnews);
}